// CrossTransformer_score1_37520834298098
// MI455X (gfx1250) — compile-verified
//
#include <hip/hip_runtime.h>
#include <hip/hip_bf16.h>

// ---------------------------------------------------------------------------
// CrossTransformer score head for MI455X (gfx1250, wave32).
// Pipeline (all on `stream`, dependencies via stream order):
//   1. mean_kernel     : per-channel fp64 mean over 1225 rows
//   2. diff_kernel     : diff[row][c] = sf - mean  (fp64, row-major)
//   3. cov_kernel      : cov = diff^T diff / 1224  (fp64)
//   4. cholesky_kernel : in-place lower Cholesky + logdet (single WG)
//   5. solve_kernel    : wave-per-column forward substitution -> probs
//   6. normsig_kernel  : L2-normalize probs, sigmoid -> sig[1225]
//   7. sw_kernel       : supports_w = sf * sig  (fp32, channels-first)
//   8. conv1x1_wmma    : qq/qv/sk/sv GEMMs via v_wmma_f32_16x16x4_f32
//   9. sim_wmma        : sim = qq^T x sk   (WMMA, K=128)
//  10. softmax_kernel  : softmax over 245, zero-padded to 248
//  11. out_wmma        : out = attn x sv   (WMMA, K=248 padded)
//  12. euclid_kernel   : -||qv - out||^2 / 49 per n  -> d_out[5]
// fp64 statistics stay on VALU (no fp64 WMMA exists on CDNA5).
// ---------------------------------------------------------------------------

typedef float v2f __attribute__((ext_vector_type(2)));
typedef float v8f __attribute__((ext_vector_type(8)));

#define C_DIM   512
#define CK_DIM  128
#define HW      49
#define NKS     25     // total supports
#define NWAY    5      // n
#define KSHOT   5      // k
#define NROWS   1225   // 25*49
#define KQ      245    // KSHOT*HW
#define KQP     248    // KQ padded to multiple of 4 for WMMA K-steps
#define LOG2PI  1.8378770664093455
#define SCALE   0.08838834764831845f   // 128^-0.5

// ---- workspace layout (bytes) ----------------------------------------------
constexpr size_t OFF_MEAN   = 0;                                     // double[512]
constexpr size_t OFF_DIFF   = OFF_MEAN  + (size_t)C_DIM * 8;         // double[1225*512]
constexpr size_t OFF_COV    = OFF_DIFF  + (size_t)NROWS * C_DIM * 8; // double[512*512] (becomes L)
constexpr size_t OFF_LOGDET = OFF_COV   + (size_t)C_DIM * C_DIM * 8; // double[1]
constexpr size_t OFF_PROBS  = OFF_LOGDET + 8;                        // double[1225]
constexpr size_t OFF_SIG    = OFF_PROBS + (size_t)NROWS * 8;         // double[1225]
constexpr size_t OFF_SW     = OFF_SIG   + (size_t)NROWS * 8;         // float[25*512*49]
constexpr size_t OFF_QQ     = OFF_SW + (size_t)NKS * C_DIM * HW * 4; // float[128*49]
constexpr size_t OFF_QV     = OFF_QQ + (size_t)CK_DIM * HW * 4;
constexpr size_t OFF_SK     = OFF_QV + (size_t)CK_DIM * HW * 4;      // float[25*128*49]
constexpr size_t OFF_SV     = OFF_SK + (size_t)NKS * CK_DIM * HW * 4;
constexpr size_t OFF_OUT    = OFF_SV + (size_t)NKS * CK_DIM * HW * 4; // float[5*128*49]
constexpr size_t OFF_SIM    = OFF_OUT + (size_t)NWAY * CK_DIM * HW * 4; // float[5*49*245]
constexpr size_t OFF_ATTN   = OFF_SIM + (size_t)NWAY * HW * KQ * 4;  // float[5*49*248]

// ---- 1. per-channel mean ---------------------------------------------------
__global__ void mean_kernel(const float* __restrict__ sup, double* __restrict__ mean) {
  int c = blockIdx.x * 256 + threadIdx.x;
  if (c >= C_DIM) return;
  double s = 0.0;
  for (int sidx = 0; sidx < NKS; ++sidx)
    for (int p = 0; p < HW; ++p)
      s += (double)sup[(size_t)sidx * C_DIM * HW + (size_t)c * HW + p];
  mean[c] = s / (double)NROWS;
}

// ---- 2. centered features (row-major fp64) ---------------------------------
__global__ void diff_kernel(const float* __restrict__ sup, const double* __restrict__ mean,
                            double* __restrict__ diff) {
  size_t idx = (size_t)blockIdx.x * 256 + threadIdx.x;
  if (idx >= (size_t)NROWS * C_DIM) return;
  int row = (int)(idx / C_DIM), c = (int)(idx % C_DIM);
  int sidx = row / HW, p = row % HW;
  diff[idx] = (double)sup[(size_t)sidx * C_DIM * HW + (size_t)c * HW + p] - mean[c];
}

// ---- 3. covariance (fp64) --------------------------------------------------
__global__ void cov_kernel(const double* __restrict__ diff, double* __restrict__ cov) {
  int i = blockIdx.x * 16 + (threadIdx.x & 15);
  int j = blockIdx.y * 16 + (threadIdx.x >> 4);
  double s = 0.0;
  for (int r = 0; r < NROWS; ++r)
    s += diff[(size_t)r * C_DIM + i] * diff[(size_t)r * C_DIM + j];
  cov[(size_t)j * C_DIM + i] = s / (double)(NROWS - 1);
}

// ---- 4. in-place lower Cholesky + logdet (single workgroup) ----------------
__global__ void cholesky_kernel(double* __restrict__ A, double* __restrict__ logdetp) {
  const int tid = threadIdx.x;  // 256 threads
  for (int k = 0; k < C_DIM; ++k) {
    if (tid == 0) A[(size_t)k * C_DIM + k] = sqrt(A[(size_t)k * C_DIM + k]);
    __syncthreads();
    double dkk = A[(size_t)k * C_DIM + k];
    for (int i = k + 1 + tid; i < C_DIM; i += 256) A[(size_t)i * C_DIM + k] /= dkk;
    __syncthreads();
    for (int i = k + 1 + tid; i < C_DIM; i += 256) {
      double lik = A[(size_t)i * C_DIM + k];
      double* Ai = A + (size_t)i * C_DIM;
      for (int j = k + 1; j <= i; ++j) Ai[j] -= lik * A[(size_t)j * C_DIM + k];
    }
    __syncthreads();
  }
  __shared__ double red[256];
  double s = 0.0;
  for (int c = tid; c < C_DIM; c += 256) s += log(A[(size_t)c * (C_DIM + 1)]);
  red[tid] = s; __syncthreads();
  for (int off = 128; off > 0; off >>= 1) {
    if (tid < off) red[tid] += red[tid + off];
    __syncthreads();
  }
  if (tid == 0) *logdetp = 2.0 * red[0];
}

// ---- 5. forward substitution, one wave per RHS column ----------------------
__global__ void solve_kernel(const double* __restrict__ L, const double* __restrict__ diff,
                             const double* __restrict__ logdetp, double* __restrict__ probs) {
  __shared__ double ys[8][C_DIM];           // 8 waves * 4KB = 32KB
  const int wave = threadIdx.x >> 5;
  const int lane = threadIdx.x & 31;
  const int gwave = blockIdx.x * 8 + wave;
  const int nwaves = gridDim.x * 8;
  volatile double* y = ys[wave];
  const double logdet = *logdetp;
  for (int col = gwave; col < NROWS; col += nwaves) {
    const double* d = diff + (size_t)col * C_DIM;
    double maha = 0.0;
    for (int i = 0; i < C_DIM; ++i) {
      const double* Lr = L + (size_t)i * C_DIM;
      double s = 0.0;
      for (int j = lane; j < i; j += 32) s += Lr[j] * y[j];
      for (int off = 16; off > 0; off >>= 1) s += __shfl_down(s, off, 32);
      double stot = __shfl(s, 0, 32);
      double yi = (d[i] - stot) / Lr[i];
      if (lane == 0) y[i] = yi;             // wave-ordered DS write
      maha += yi * yi;                      // identical on all lanes
    }
    if (lane == 0)
      probs[col] = exp(-0.5 * ((double)C_DIM * LOG2PI + logdet + maha));
  }
}

// ---- 6. normalize + sigmoid (single workgroup) -----------------------------
__global__ void normsig_kernel(const double* __restrict__ probs, double* __restrict__ sig) {
  __shared__ double red[256];
  int tid = threadIdx.x;
  double s = 0.0;
  for (int i = tid; i < NROWS; i += 256) { double p = probs[i]; s += p * p; }
  red[tid] = s; __syncthreads();
  for (int off = 128; off > 0; off >>= 1) {
    if (tid < off) red[tid] += red[tid + off];
    __syncthreads();
  }
  double norm = sqrt(red[0]);
  if (norm < 1e-12) norm = 1e-12;
  __syncthreads();
  for (int i = tid; i < NROWS; i += 256)
    sig[i] = 1.0 / (1.0 + exp(-(probs[i] / norm)));
}

// ---- 7. weighted supports (fp32 channels-first) ----------------------------
__global__ void sw_kernel(const double* __restrict__ diff, const double* __restrict__ mean,
                          const double* __restrict__ sig, float* __restrict__ SW) {
  size_t idx = (size_t)blockIdx.x * 256 + threadIdx.x;
  if (idx >= (size_t)NROWS * C_DIM) return;
  int row = (int)(idx / C_DIM), c = (int)(idx % C_DIM);
  int sidx = row / HW, p = row % HW;
  double sf = diff[idx] + mean[c];
  SW[(size_t)sidx * C_DIM * HW + (size_t)c * HW + p] = (float)(sf * sig[row]);
}

// ---- 8. 1x1 convs as GEMMs via V_WMMA_F32_16X16X4_F32 ----------------------
// grid: x = mtile(8)*ntile(4) = 32, y = image (0=query, 1..25 = support s),
//       z = weight select (0 = W_qk, 1 = W_v).  blockDim = 32 (one wave).
// Out-of-range output columns use a clamped address (dup of col 48) so all
// loads are unconditional; stores are guarded, columns are lane-private.
__global__ void conv1x1_wmma(const float* __restrict__ Wqk, const float* __restrict__ Wv,
                             const float* __restrict__ query, const float* __restrict__ SW,
                             float* __restrict__ qq, float* __restrict__ qv,
                             float* __restrict__ sk, float* __restrict__ sv) {
  const int mt   = blockIdx.x >> 2;
  const int nt   = blockIdx.x & 3;
  const int img  = blockIdx.y;
  const int wsel = blockIdx.z;
  const float* Wm = wsel ? Wv : Wqk;                         // 128 x 512 row-major
  const float* X  = (img == 0) ? query : (SW + (size_t)(img - 1) * C_DIM * HW);
  float* dst = (img == 0) ? (wsel ? qv : qq)
                          : ((wsel ? sv : sk) + (size_t)(img - 1) * CK_DIM * HW);
  const int lane = threadIdx.x & 31;
  const int half = lane >> 4;      // selects K pair within fragment
  const int l16  = lane & 15;
  const int m0 = mt * 16;
  const int p  = nt * 16 + l16;              // output pixel column (pad 49 -> 64)
  const int pc = (p < HW) ? p : (HW - 1);    // clamped -> unconditional loads
  const float* Wr = Wm + (size_t)(m0 + l16) * C_DIM;
  const float* Xc = X + pc;

  v8f acc = {};
  for (int k = 0; k < C_DIM; k += 4) {
    const int ka = k + 2 * half;
    v2f a, b;
    a.x = Wr[ka];
    a.y = Wr[ka + 1];
    b.x = Xc[(size_t)ka * HW];
    b.y = Xc[(size_t)(ka + 1) * HW];
    acc = __builtin_amdgcn_wmma_f32_16x16x4_f32(false, a, false, b,
                                                (short)0, acc, false, false);
  }
  if (p < HW) {
#pragma unroll
    for (int r = 0; r < 8; ++r) {
      int m = m0 + r + 8 * half;
      dst[(size_t)m * HW + p] = acc[r];
    }
  }
}

// ---- 9. sim = qq^T x sk via WMMA (per n: 49x245, K=128) --------------------
// grid: x = p tile (4), y = kq tile (16), z = n (5). blockDim = 32.
__global__ void sim_wmma(const float* __restrict__ qq, const float* __restrict__ sk,
                         float* __restrict__ simw) {
  const int mt = blockIdx.x, nt = blockIdx.y, nidx = blockIdx.z;
  const int lane = threadIdx.x & 31;
  const int half = lane >> 4;
  const int l16  = lane & 15;
  const int m  = mt * 16 + l16;                        // query pixel p
  const int mA = (m < HW) ? m : (HW - 1);
  const int nq = nt * 16 + l16;                        // kq = k*49 + ij
  const int nqc = (nq < KQ) ? nq : (KQ - 1);
  const int kk = nqc / HW, ij = nqc % HW;
  const float* qb  = qq + mA;
  const float* skb = sk + ((size_t)(nidx * KSHOT + kk) * CK_DIM) * HW + ij;

  v8f acc = {};
  for (int k = 0; k < CK_DIM; k += 4) {
    const int ka = k + 2 * half;
    v2f a, b;
    a.x = qb[(size_t)ka * HW];
    a.y = qb[(size_t)(ka + 1) * HW];
    b.x = skb[(size_t)ka * HW];
    b.y = skb[(size_t)(ka + 1) * HW];
    acc = __builtin_amdgcn_wmma_f32_16x16x4_f32(false, a, false, b,
                                                (short)0, acc, false, false);
  }
  if (nq < KQ) {
#pragma unroll
    for (int r = 0; r < 8; ++r) {
      int p = mt * 16 + r + 8 * half;
      if (p < HW) simw[((size_t)nidx * HW + p) * KQ + nq] = acc[r] * SCALE;
    }
  }
}

// ---- 10. softmax over kq (245), zero-padded to 248 -------------------------
__global__ void softmax_kernel(const float* __restrict__ simw, float* __restrict__ attnw) {
  const int nidx = blockIdx.x / HW;
  const int p    = blockIdx.x % HW;
  const int tid  = threadIdx.x;        // 256
  const float* row = simw + ((size_t)nidx * HW + p) * KQ;
  float* arow = attnw + ((size_t)nidx * HW + p) * KQP;
  __shared__ float red[256];
  float v = (tid < KQ) ? row[tid] : -INFINITY;
  red[tid] = v; __syncthreads();
  for (int off = 128; off > 0; off >>= 1) {
    if (tid < off) red[tid] = fmaxf(red[tid], red[tid + off]);
    __syncthreads();
  }
  const float m = red[0];
  __syncthreads();
  float e = (tid < KQ) ? expf(v - m) : 0.0f;
  red[tid] = e; __syncthreads();
  for (int off = 128; off > 0; off >>= 1) {
    if (tid < off) red[tid] += red[tid + off];
    __syncthreads();
  }
  const float inv = 1.0f / red[0];
  if (tid < KQP) arow[tid] = (tid < KQ) ? e * inv : 0.0f;  // exact zero padding
}

// ---- 11. out = attn x sv via WMMA (per n: 49x128, K=248 padded) ------------
// grid: x = p tile (4), y = c tile (8), z = n (5). blockDim = 32.
__global__ void out_wmma(const float* __restrict__ attnw, const float* __restrict__ sv,
                         float* __restrict__ out) {
  const int mt = blockIdx.x, nt = blockIdx.y, nidx = blockIdx.z;
  const int lane = threadIdx.x & 31;
  const int half = lane >> 4;
  const int l16  = lane & 15;
  const int m  = mt * 16 + l16;                       // query pixel p
  const int mA = (m < HW) ? m : (HW - 1);
  const int c  = nt * 16 + l16;                       // output channel, always < 128
  const float* arow = attnw + ((size_t)nidx * HW + mA) * KQP;
  const float* svb  = sv + (size_t)nidx * KSHOT * CK_DIM * HW + (size_t)c * HW;

  v8f acc = {};
  for (int k = 0; k < KQP; k += 4) {
    const int ka = k + 2 * half;
    const int k0 = (ka < KQ) ? ka : (KQ - 1);         // attn pad is 0 -> b irrelevant
    const int k1 = (ka + 1 < KQ) ? (ka + 1) : (KQ - 1);
    v2f a, b;
    a.x = arow[ka];
    a.y = arow[ka + 1];
    b.x = svb[(size_t)(k0 / HW) * CK_DIM * HW + (k0 % HW)];
    b.y = svb[(size_t)(k1 / HW) * CK_DIM * HW + (k1 % HW)];
    acc = __builtin_amdgcn_wmma_f32_16x16x4_f32(false, a, false, b,
                                                (short)0, acc, false, false);
  }
#pragma unroll
  for (int r = 0; r < 8; ++r) {
    int p = mt * 16 + r + 8 * half;
    if (p < HW) out[((size_t)nidx * CK_DIM + c) * HW + p] = acc[r];
  }
}

// ---- 12. negative mean squared distance per n ------------------------------
__global__ void euclid_kernel(const float* __restrict__ qv, const float* __restrict__ out,
                              float* __restrict__ res) {
  const int nidx = blockIdx.x;
  __shared__ float red[256];
  float s = 0.0f;
  for (int i = threadIdx.x; i < CK_DIM * HW; i += 256) {
    float d = qv[i] - out[(size_t)nidx * CK_DIM * HW + i];
    s += d * d;
  }
  red[threadIdx.x] = s; __syncthreads();
  for (int off = 128; off > 0; off >>= 1) {
    if (threadIdx.x < off) red[threadIdx.x] += red[threadIdx.x + off];
    __syncthreads();
  }
  if (threadIdx.x == 0) res[nidx] = -(red[0] / (float)HW);
}

// ---------------------------------------------------------------------------
extern "C" void kernel_launch(void* const* d_in, const int* in_sizes, int n_in,
                              void* d_out, int out_size, void* d_ws, size_t ws_size,
                              hipStream_t stream) {
  const float* query = (const float*)d_in[0];   // (1,512,7,7)
  const float* sup   = (const float*)d_in[1];   // (25,512,7,7)
  const float* Wqk   = (const float*)d_in[2];   // (128,512)
  const float* Wv    = (const float*)d_in[3];   // (128,512)
  float* res = (float*)d_out;                   // (1,5)

  char* ws = (char*)d_ws;
  double* mean   = (double*)(ws + OFF_MEAN);
  double* diff   = (double*)(ws + OFF_DIFF);
  double* cov    = (double*)(ws + OFF_COV);
  double* logdet = (double*)(ws + OFF_LOGDET);
  double* probs  = (double*)(ws + OFF_PROBS);
  double* sig    = (double*)(ws + OFF_SIG);
  float*  SW     = (float*)(ws + OFF_SW);
  float*  qq     = (float*)(ws + OFF_QQ);
  float*  qv     = (float*)(ws + OFF_QV);
  float*  sk     = (float*)(ws + OFF_SK);
  float*  sv     = (float*)(ws + OFF_SV);
  float*  outb   = (float*)(ws + OFF_OUT);
  float*  simw   = (float*)(ws + OFF_SIM);
  float*  attnw  = (float*)(ws + OFF_ATTN);

  const int elems = NROWS * C_DIM;
  const int eb = (elems + 255) / 256;

  mean_kernel<<<dim3(2), dim3(256), 0, stream>>>(sup, mean);
  diff_kernel<<<dim3(eb), dim3(256), 0, stream>>>(sup, mean, diff);
  cov_kernel<<<dim3(32, 32), dim3(256), 0, stream>>>(diff, cov);
  cholesky_kernel<<<dim3(1), dim3(256), 0, stream>>>(cov, logdet);
  solve_kernel<<<dim3(40), dim3(256), 0, stream>>>(cov, diff, logdet, probs);
  normsig_kernel<<<dim3(1), dim3(256), 0, stream>>>(probs, sig);
  sw_kernel<<<dim3(eb), dim3(256), 0, stream>>>(diff, mean, sig, SW);
  conv1x1_wmma<<<dim3(32, NKS + 1, 2), dim3(32), 0, stream>>>(Wqk, Wv, query, SW,
                                                              qq, qv, sk, sv);
  sim_wmma<<<dim3(4, 16, NWAY), dim3(32), 0, stream>>>(qq, sk, simw);
  softmax_kernel<<<dim3(NWAY * HW), dim3(256), 0, stream>>>(simw, attnw);
  out_wmma<<<dim3(4, 8, NWAY), dim3(32), 0, stream>>>(attnw, sv, outb);
  euclid_kernel<<<dim3(NWAY), dim3(256), 0, stream>>>(qv, outb, res);
}